// AdditiveAttention_16836271800570
// MI455X (gfx1250) — compile-verified
//
#include <hip/hip_runtime.h>

typedef __attribute__((ext_vector_type(2))) float v2f;
typedef __attribute__((ext_vector_type(8))) float v8f;

#define B_   8
#define NQ_  1024
#define NK_  1024
#define DQ_  256
#define DV_  256
#define H_   64
#define NEG_INF_ (-1e20f)

// CDNA5 hardware tanh (TRANS op). Batch 4 independent tanh back-to-back so each
// trans op is followed by an independent instruction, then a v_nop before the
// last result can be consumed (ISA §7.4.3 TRANS hazard rule).
__device__ __forceinline__ void tanh4(float a0, float a1, float a2, float a3,
                                      float& t0, float& t1, float& t2, float& t3) {
  asm volatile("v_tanh_f32 %0, %4\n\t"
               "v_tanh_f32 %1, %5\n\t"
               "v_tanh_f32 %2, %6\n\t"
               "v_tanh_f32 %3, %7\n\t"
               "v_nop"
               : "=&v"(t0), "=&v"(t1), "=&v"(t2), "=&v"(t3)
               : "v"(a0), "v"(a1), "v"(a2), "v"(a3));
}

// Y[b, r, h] = sum_d X[b, r, d] * W[d, h]   (X: (B,1024,256), W: (256,64))
// One wave per 16x16 output tile, f32 WMMA 16x16x4 over the 256-deep K dim.
__global__ __launch_bounds__(32) void AdditiveAttention_proj_kernel(
    const float* __restrict__ X, const float* __restrict__ W,
    float* __restrict__ Y) {
  int tile = blockIdx.x;
  int ht = tile & 3;           // H/16 = 4 tiles
  int rt = (tile >> 2) & 63;   // NQ/16 = 64 tiles
  int b  = tile >> 8;
  const float* Xb = X + (size_t)b * NQ_ * DQ_;
  float* Yb = Y + (size_t)b * NQ_ * H_;

  int lane = threadIdx.x & 31;
  int m    = lane & 15;        // M row (A) / N col (B,D)
  int half = lane >> 4;        // K split across lane halves
  int r0 = rt * 16, h0 = ht * 16;

  v8f acc = {};
  for (int kk = 0; kk < DQ_; kk += 4) {
    int ko = kk + (half << 1);
    // A: 16x4 f32 tile — v0 holds K, v1 holds K+1 (per lane half)
    v2f a = *(const v2f*)(Xb + (size_t)(r0 + m) * DQ_ + ko);
    // B: 4x16 f32 tile — row K across lanes within a VGPR
    const float* bp = W + (size_t)ko * H_ + h0 + m;
    v2f bm;
    bm.x = bp[0];
    bm.y = bp[H_];
    acc = __builtin_amdgcn_wmma_f32_16x16x4_f32(false, a, false, bm,
                                                (short)0, acc, false, false);
  }
  // C/D layout: VGPR j -> row j (lanes 0-15), row j+8 (lanes 16-31)
  for (int j = 0; j < 8; ++j) {
    int row = r0 + j + half * 8;
    Yb[(size_t)row * H_ + h0 + m] = acc[j];
  }
}

// Fused scores + masked softmax + P@V for a 16-query tile of one batch.
__global__ __launch_bounds__(256) void AdditiveAttention_attn_kernel(
    const float* __restrict__ Qh, const float* __restrict__ Kh,
    const float* __restrict__ Wv, const float* __restrict__ V,
    const int* __restrict__ valid_lens, float* __restrict__ out) {
  __shared__ float sQh[16][68];     // padded (68 mod 64 banks = 4)
  __shared__ float sKh[16][68];
  __shared__ float sWv[64];
  __shared__ float sS[16][1028];    // scores -> exp(scores), padded stride
  __shared__ float sRed[16][17];
  __shared__ float sRowMax[16];
  __shared__ float sRowInv[16];

  int bid = blockIdx.x;
  int qt = bid & 63;
  int b  = bid >> 6;
  int q0 = qt * 16;
  int tid = threadIdx.x;

  const float* Qhb = Qh + ((size_t)b * NQ_ + q0) * H_;
  const float* Khb = Kh + (size_t)b * NK_ * H_;
  const float* Vb  = V  + (size_t)b * NK_ * DV_;
  int vlen = valid_lens[b];

  // Stage Qh tile (16x64) and Wv
  {
    int r = tid >> 4, c4 = (tid & 15) * 4;
    float4 t = *(const float4*)(Qhb + (size_t)r * H_ + c4);
    *(float4*)&sQh[r][c4] = t;
    if (tid < 64) sWv[tid] = Wv[tid];
  }
  __syncthreads();

  // ---- Phase 1: scores[q][k] = sum_h tanh(Qh+Kh)*Wv, masked ----
  int qq = tid & 15;       // query row
  int kc = tid >> 4;       // key row within 16-row chunk
  for (int k0 = 0; k0 < NK_; k0 += 16) {
    {
      int r = tid >> 4, c4 = (tid & 15) * 4;
      float4 t = *(const float4*)(Khb + (size_t)(k0 + r) * H_ + c4);
      *(float4*)&sKh[r][c4] = t;
    }
    __syncthreads();

    float accs = 0.f;
    const float4* qrow = (const float4*)&sQh[qq][0];
    const float4* krow = (const float4*)&sKh[kc][0];
    const float4* wrow = (const float4*)&sWv[0];
    #pragma unroll 4
    for (int h4 = 0; h4 < 16; ++h4) {
      float4 qv = qrow[h4];
      float4 kv = krow[h4];
      float4 wv = wrow[h4];
      float t0, t1, t2, t3;
      tanh4(qv.x + kv.x, qv.y + kv.y, qv.z + kv.z, qv.w + kv.w, t0, t1, t2, t3);
      accs = fmaf(t0, wv.x, accs);
      accs = fmaf(t1, wv.y, accs);
      accs = fmaf(t2, wv.z, accs);
      accs = fmaf(t3, wv.w, accs);
    }
    int col = k0 + kc;
    sS[qq][col] = (col >= vlen) ? NEG_INF_ : accs;
    __syncthreads();
  }

  // ---- Phase 2: masked softmax (keep unnormalized exp, fold 1/sum later) ----
  int row = tid >> 4, sub = tid & 15;
  float mx = NEG_INF_;
  for (int c = sub; c < NK_; c += 16) mx = fmaxf(mx, sS[row][c]);
  sRed[row][sub] = mx;
  __syncthreads();
  if (sub == 0) {
    float mm = sRed[row][0];
    for (int i = 1; i < 16; ++i) mm = fmaxf(mm, sRed[row][i]);
    sRowMax[row] = mm;
  }
  __syncthreads();
  float rmax = sRowMax[row];
  float ssum = 0.f;
  for (int c = sub; c < NK_; c += 16) {
    float e = __expf(sS[row][c] - rmax);
    sS[row][c] = e;
    ssum += e;
  }
  sRed[row][sub] = ssum;
  __syncthreads();
  if (sub == 0) {
    float tt = 0.f;
    for (int i = 0; i < 16; ++i) tt += sRed[row][i];
    sRowInv[row] = 1.0f / tt;
  }
  __syncthreads();

  // ---- Phase 3: out = (P @ V) * rowInv via f32 WMMA 16x16x4 ----
  int wave = tid >> 5;
  int lane = tid & 31;
  int c0 = wave * 32;          // each wave owns 32 of the 256 DV columns
  int mrow = lane & 15;
  int half = lane >> 4;

  v8f acc0 = {}, acc1 = {};
  for (int k0 = 0; k0 < NK_; k0 += 4) {
    int ko = k0 + (half << 1);
    v2f a = *(const v2f*)&sS[mrow][ko];            // A: P tile 16x4
    const float* vp = Vb + (size_t)ko * DV_ + c0 + mrow;
    v2f b0; b0.x = vp[0];  b0.y = vp[DV_];         // B: V tile 4x16 (cols c0..)
    v2f b1; b1.x = vp[16]; b1.y = vp[DV_ + 16];    // B: cols c0+16..
    acc0 = __builtin_amdgcn_wmma_f32_16x16x4_f32(false, a, false, b0,
                                                 (short)0, acc0, false, false);
    acc1 = __builtin_amdgcn_wmma_f32_16x16x4_f32(false, a, false, b1,
                                                 (short)0, acc1, false, false);
  }

  float* ob = out + ((size_t)b * NQ_ + q0) * DV_;
  for (int j = 0; j < 8; ++j) {
    int r = j + half * 8;
    float sc = sRowInv[r];
    ob[(size_t)r * DV_ + c0 + mrow]      = acc0[j] * sc;
    ob[(size_t)r * DV_ + c0 + 16 + mrow] = acc1[j] * sc;
  }
}

extern "C" void kernel_launch(void* const* d_in, const int* in_sizes, int n_in,
                              void* d_out, int out_size, void* d_ws, size_t ws_size,
                              hipStream_t stream) {
  (void)in_sizes; (void)n_in; (void)out_size; (void)ws_size;
  const float* Q  = (const float*)d_in[0];
  const float* K  = (const float*)d_in[1];
  const float* V  = (const float*)d_in[2];
  const float* Wq = (const float*)d_in[3];
  const float* Wk = (const float*)d_in[4];
  const float* Wv = (const float*)d_in[5];
  const int*   vl = (const int*)d_in[6];
  float* out = (float*)d_out;

  float* Qh = (float*)d_ws;                         // (B, NQ, H)
  float* Kh = Qh + (size_t)B_ * NQ_ * H_;           // (B, NK, H)

  // Projections: B * (1024/16) * (64/16) = 2048 tiles each, 1 wave per tile
  AdditiveAttention_proj_kernel<<<B_ * 64 * 4, 32, 0, stream>>>(Q, Wq, Qh);
  AdditiveAttention_proj_kernel<<<B_ * 64 * 4, 32, 0, stream>>>(K, Wk, Kh);

  // Fused scores + softmax + P@V: one block per (batch, 16-query tile)
  AdditiveAttention_attn_kernel<<<B_ * 64, 256, 0, stream>>>(Qh, Kh, Wv, V, vl, out);
}